// Mind2D_85976655331979
// MI455X (gfx1250) — compile-verified
//
#include <hip/hip_runtime.h>
#include <cstdint>
#include <math.h>

#define AS1 __attribute__((address_space(1)))
#define AS3 __attribute__((address_space(3)))

#ifndef __has_builtin
#define __has_builtin(x) 0
#endif

// ---------------------------------------------------------------------------
// CDNA5 async global->LDS copy (per-lane gather into LDS, tracked by ASYNCcnt)
// ---------------------------------------------------------------------------
__device__ __forceinline__ void async_copy_f32_to_lds(const float* g, float* l) {
#if __has_builtin(__builtin_amdgcn_global_load_async_to_lds_b32)
    __builtin_amdgcn_global_load_async_to_lds_b32((AS1 int*)g, (AS3 int*)l, 0, 0);
#else
    asm volatile("global_load_async_to_lds_b32 %0, %1, off"
                 :: "v"((AS3 float*)l), "v"(g)
                 : "memory");
#endif
}

__device__ __forceinline__ void wait_async_zero() {
#if __has_builtin(__builtin_amdgcn_s_wait_asynccnt)
    __builtin_amdgcn_s_wait_asynccnt(0);
#else
    asm volatile("s_wait_asynccnt 0x0" ::: "memory");
#endif
}

// ---------------------------------------------------------------------------
// Problem constants
// ---------------------------------------------------------------------------
#define BATCH 8
#define H     1024
#define W     1024
#define OH    1026           // output spatial size
#define TO    32             // output tile
#define XT    40             // padded-input tile (TO + 8)
#define SQT   36             // squared-diff tile (TO + 4)

// ---------------------------------------------------------------------------
// Shared tile body. P(u,v) = zero-pad(edge-pad(xn,2),1) over 1030x1030:
//   P(u,v) = 0 if u or v on the outer ring, else xn[clamp(u-3),clamp(v-3)].
// d0 = P(a,b+2)-P(a+2,b); d1 = P(a+2,b+4)-P(a,b+2);
// d2 = P(a+4,b+2)-P(a+2,b); d3 = P(a+4,b+2)-P(a+2,b+4).
// ssd_c(i,j) = (1/25) * sum_{d in [-2,2]^2} d_c(clamp(i+di),clamp(j+dj))^2.
// WRITE=false: per-block sum of mind_var.  WRITE=true: emit exp(-mind/var).
// ---------------------------------------------------------------------------
template <bool WRITE>
__device__ __forceinline__
void mind_tile_body(const float* __restrict__ x, const float* __restrict__ scal,
                    float* __restrict__ out, float* __restrict__ partial) {
    __shared__ float s_xp[XT * XT];
    __shared__ float s_sq[4][SQT * SQT];
    __shared__ float s_red[256];

    const int tid = threadIdx.x;
    const int b   = blockIdx.z;
    const int ti  = blockIdx.y * TO;
    const int tj  = blockIdx.x * TO;

    const float* xb = x + (size_t)b * H * W;

    // Phase 1: async gather of raw x (replicate-clamp folded into addressing)
    for (int e = tid; e < XT * XT; e += 256) {
        int k = e / XT, l = e - k * XT;
        int u = ti - 2 + k, v = tj - 2 + l;
        int xr = u - 3; xr = xr < 0 ? 0 : (xr > H - 1 ? H - 1 : xr);
        int xc = v - 3; xc = xc < 0 ? 0 : (xc > W - 1 ? W - 1 : xc);
        async_copy_f32_to_lds(xb + (size_t)xr * W + xc, &s_xp[e]);
    }
    const float vmin = scal[0];
    const float inv  = scal[1];
    wait_async_zero();
    __syncthreads();

    // Phase 2: normalize + zero outer conv ring, in place
    for (int e = tid; e < XT * XT; e += 256) {
        int k = e / XT, l = e - k * XT;
        int u = ti - 2 + k, v = tj - 2 + l;
        bool ring = (u == 0) | (u == 1029) | (v == 0) | (v == 1029);
        float val = (s_xp[e] - vmin) * inv;
        s_xp[e] = ring ? 0.0f : val;
    }
    __syncthreads();

    // Phase 3: squared shift-differences, 4 channels
    for (int e = tid; e < SQT * SQT; e += 256) {
        int r = e / SQT, s = e - r * SQT;
        const float* p = &s_xp[r * XT + s];
        float tA = p[0 * XT + 2];   // (a,   b+2)
        float tB = p[2 * XT + 0];   // (a+2, b)
        float tC = p[2 * XT + 4];   // (a+2, b+4)
        float tD = p[4 * XT + 2];   // (a+4, b+2)
        float d0 = tA - tB;
        float d1 = tC - tA;
        float d2 = tD - tB;
        float d3 = tD - tC;
        s_sq[0][e] = d0 * d0;
        s_sq[1][e] = d1 * d1;
        s_sq[2][e] = d2 * d2;
        s_sq[3][e] = d3 * d3;
    }
    __syncthreads();

    // Phase 4: clamped 5x5 box sums + epilogue
    float mvsum = 0.0f;
    float lo = 0.0f, hi = 0.0f;
    if (WRITE) {
        float mvm = scal[2];
        lo = mvm * 0.001f;
        hi = mvm * 1000.0f;
    }

    for (int e = tid; e < TO * TO; e += 256) {
        int py = e / TO, px = e - py * TO;
        int i = ti + py, j = tj + px;
        if (i >= OH || j >= OH) continue;

        int rr[5], cc[5];
#pragma unroll
        for (int d = 0; d < 5; ++d) {
            int a = i + d - 2; a = a < 0 ? 0 : (a > OH - 1 ? OH - 1 : a);
            rr[d] = a - ti + 2;
            int c2 = j + d - 2; c2 = c2 < 0 ? 0 : (c2 > OH - 1 ? OH - 1 : c2);
            cc[d] = c2 - tj + 2;
        }

        float ssd[4];
#pragma unroll
        for (int c = 0; c < 4; ++c) {
            float acc = 0.0f;
#pragma unroll
            for (int dy = 0; dy < 5; ++dy) {
                const float* rp = &s_sq[c][rr[dy] * SQT];
                acc += rp[cc[0]] + rp[cc[1]] + rp[cc[2]] + rp[cc[3]] + rp[cc[4]];
            }
            ssd[c] = acc * 0.04f;  // / 25
        }

        float m  = fminf(fminf(ssd[0], ssd[1]), fminf(ssd[2], ssd[3]));
        float mv = (ssd[0] + ssd[1] + ssd[2] + ssd[3]) * 0.25f - m;

        if (WRITE) {
            float mvc  = fminf(fmaxf(mv, lo), hi);
            float rinv = 1.0f / mvc;
            size_t base = ((size_t)b * 4) * (size_t)(OH * OH) + (size_t)i * OH + j;
            // streaming output: non-temporal stores keep x resident in L2
#pragma unroll
            for (int c = 0; c < 4; ++c)
                __builtin_nontemporal_store(expf(-(ssd[c] - m) * rinv),
                                            &out[base + (size_t)c * (OH * OH)]);
        } else {
            mvsum += mv;
        }
    }

    if (!WRITE) {
        s_red[tid] = mvsum;
        __syncthreads();
        for (int s = 128; s > 0; s >>= 1) {
            if (tid < s) s_red[tid] += s_red[tid + s];
            __syncthreads();
        }
        if (tid == 0) {
            int bidx = (blockIdx.z * gridDim.y + blockIdx.y) * gridDim.x + blockIdx.x;
            partial[bidx] = s_red[0];
        }
    }
}

// Concrete kernels (defined first so the tile body shows up at the top of the
// device assembly, async-to-LDS ops included).
__global__ __launch_bounds__(256)
void mind_tile_write(const float* __restrict__ x, const float* __restrict__ scal,
                     float* __restrict__ out) {
    mind_tile_body<true>(x, scal, out, nullptr);
}

__global__ __launch_bounds__(256)
void mind_tile_reduce(const float* __restrict__ x, const float* __restrict__ scal,
                      float* __restrict__ partial) {
    mind_tile_body<false>(x, scal, nullptr, partial);
}

// ---------------------------------------------------------------------------
// Pass 1a: per-block min/max of x
// ---------------------------------------------------------------------------
__global__ __launch_bounds__(256)
void mind_minmax_part(const float* __restrict__ x, float* __restrict__ wmin,
                      float* __restrict__ wmax, int n) {
    __shared__ float smin[256], smax[256];
    float lmin = __FLT_MAX__, lmax = -__FLT_MAX__;
    int stride = gridDim.x * 256;
    for (int idx = blockIdx.x * 256 + threadIdx.x; idx < n; idx += stride) {
        float v = x[idx];
        lmin = fminf(lmin, v);
        lmax = fmaxf(lmax, v);
    }
    smin[threadIdx.x] = lmin;
    smax[threadIdx.x] = lmax;
    __syncthreads();
    for (int s = 128; s > 0; s >>= 1) {
        if (threadIdx.x < s) {
            smin[threadIdx.x] = fminf(smin[threadIdx.x], smin[threadIdx.x + s]);
            smax[threadIdx.x] = fmaxf(smax[threadIdx.x], smax[threadIdx.x + s]);
        }
        __syncthreads();
    }
    if (threadIdx.x == 0) {
        wmin[blockIdx.x] = smin[0];
        wmax[blockIdx.x] = smax[0];
    }
}

// Pass 1b: final min/max -> scal[0]=vmin, scal[1]=1/(range+1e-6)
__global__ __launch_bounds__(256)
void mind_minmax_final(const float* __restrict__ wmin, const float* __restrict__ wmax,
                       float* __restrict__ scal, int nb) {
    __shared__ float smin[256], smax[256];
    float lmin = __FLT_MAX__, lmax = -__FLT_MAX__;
    for (int i = threadIdx.x; i < nb; i += 256) {
        lmin = fminf(lmin, wmin[i]);
        lmax = fmaxf(lmax, wmax[i]);
    }
    smin[threadIdx.x] = lmin;
    smax[threadIdx.x] = lmax;
    __syncthreads();
    for (int s = 128; s > 0; s >>= 1) {
        if (threadIdx.x < s) {
            smin[threadIdx.x] = fminf(smin[threadIdx.x], smin[threadIdx.x + s]);
            smax[threadIdx.x] = fmaxf(smax[threadIdx.x], smax[threadIdx.x + s]);
        }
        __syncthreads();
    }
    if (threadIdx.x == 0) {
        scal[0] = smin[0];
        scal[1] = 1.0f / (smax[0] - smin[0] + 1e-6f);
    }
}

// Pass 3: mvm = mean of mind_var over [8,1,1026,1026] -> scal[2]
__global__ __launch_bounds__(256)
void mind_mvm_final(const float* __restrict__ partial, float* __restrict__ scal, int np) {
    __shared__ float sred[256];
    float s = 0.0f;
    for (int i = threadIdx.x; i < np; i += 256) s += partial[i];
    sred[threadIdx.x] = s;
    __syncthreads();
    for (int k = 128; k > 0; k >>= 1) {
        if (threadIdx.x < k) sred[threadIdx.x] += sred[threadIdx.x + k];
        __syncthreads();
    }
    if (threadIdx.x == 0)
        scal[2] = sred[0] / ((float)BATCH * (float)OH * (float)OH);
}

// ---------------------------------------------------------------------------
// Launch: ws layout (floats):
//   [0..1023] block mins, [1024..2047] block maxs,
//   [2048]=vmin [2049]=inv_range [2050]=mvm [2051]=pad,
//   [2052..2052+8712) per-block mind_var partial sums.  (~43 KB total)
// ---------------------------------------------------------------------------
extern "C" void kernel_launch(void* const* d_in, const int* in_sizes, int n_in,
                              void* d_out, int out_size, void* d_ws, size_t ws_size,
                              hipStream_t stream) {
    const float* x   = (const float*)d_in[0];
    float*       out = (float*)d_out;
    float*       ws  = (float*)d_ws;

    float* wmin = ws;
    float* wmax = ws + 1024;
    float* scal = ws + 2048;
    float* part = ws + 2052;

    const int n = in_sizes[0];  // 8*1024*1024

    mind_minmax_part<<<1024, 256, 0, stream>>>(x, wmin, wmax, n);
    mind_minmax_final<<<1, 256, 0, stream>>>(wmin, wmax, scal, 1024);

    dim3 grid((OH + TO - 1) / TO, (OH + TO - 1) / TO, BATCH);  // 33 x 33 x 8
    mind_tile_reduce<<<grid, 256, 0, stream>>>(x, scal, part);
    mind_mvm_final<<<1, 256, 0, stream>>>(part, scal, 33 * 33 * BATCH);
    mind_tile_write<<<grid, 256, 0, stream>>>(x, scal, out);
}